// PrecNetNorm_77438260346966
// MI455X (gfx1250) — compile-verified
//
#include <hip/hip_runtime.h>

#define N_NODES 262144
#define E_EDGES (3 * N_NODES)
#define HID 32
#define FAM 64            // node families per block
#define NTHREADS 256

typedef __attribute__((ext_vector_type(2))) float v2f;
typedef __attribute__((ext_vector_type(8))) float v8f;

// ---------------- LDS layout (floats) ----------------
#define MROWS 80                    // rows covered by 5 M-tiles
#define AROWS 81                    // allocated rows (reads may touch row 80)
#define RS    34                    // row stride (even pair, odd/2 -> conflict-free WMMA A loads)
#define BUFSZ (AROWS * RS)          // 2754
#define X_O    0                    // 4 * MROWS raw input staging
#define NBUF_O (4 * MROWS)          // node features (in-place updated)
#define E_O    (NBUF_O + BUFSZ)     // e[t] = E_O + t*BUFSZ
#define SC_O   (E_O + 3 * BUFSZ)    // agg staging
#define F_NEW2 (SC_O + BUFSZ)       // packed B-fragments
#define F_EEW2 (F_NEW2 + 1024)
#define F_MPWE (F_EEW2 + 1024)
#define F_MPWN (F_MPWE + 3072)
#define F_EDW1 (F_MPWN + 2048)
#define V_NEW1 (F_EDW1 + 1024)
#define V_NEB1 (V_NEW1 + 32)
#define V_NEB2 (V_NEB1 + 32)
#define V_EEW1 (V_NEB2 + 32)
#define V_EEB1 (V_EEW1 + 32)
#define V_EEB2 (V_EEB1 + 32)
#define V_MPBE (V_EEB2 + 32)
#define V_MPBN (V_MPBE + 32)
#define V_EDB1 (V_MPBN + 32)
#define V_EDW2 (V_EDB1 + 32)
#define V_EDB2 (V_EDW2 + 32)
#define V_NORM (V_EDB2 + 1)
#define V_INV  (V_NORM + 1)
#define LDS_FLOATS (V_INV + 1)

// ---------------- WMMA helpers ----------------
__device__ __forceinline__ v8f wmma4(v2f a, v2f b, v8f c) {
    // D(16x16,f32) = A(16x4,f32) x B(4x16,f32) + C
    return __builtin_amdgcn_wmma_f32_16x16x4_f32(false, a, false, b, (short)0, c,
                                                 false, false);
}

// A fragment: lane<16 -> row=rb+lane, K=k0..k0+1 ; lane>=16 -> row=rb+lane-16, K=k0+2..k0+3
__device__ __forceinline__ v2f lafrag(const float* buf, int rb, int k0, int lane) {
    const float* p = buf + (rb + (lane & 15)) * RS + k0 + ((lane >> 4) << 1);
    v2f a; a.x = p[0]; a.y = p[1]; return a;
}

// B fragment from pre-swizzled packed storage: one aligned 8B load per lane
__device__ __forceinline__ v2f lbfrag(const float* frag, int kt, int nt, int lane) {
    const float* p = frag + ((((kt << 1) | nt) << 5) + lane) * 2;
    return *(const v2f*)p;
}

__device__ __forceinline__ v8f cbias(const float* b, int nt, int lane) {
    float v = b[(nt << 4) + (lane & 15)];
    v8f c;
#pragma unroll
    for (int g = 0; g < 8; ++g) c[g] = v;
    return c;
}

__device__ __forceinline__ void cstore(float* dst, int rb, int nt, int lane, v8f c,
                                       bool relu) {
    int col = (nt << 4) + (lane & 15);
    int r0  = rb + ((lane >> 4) << 3);
#pragma unroll
    for (int g = 0; g < 8; ++g) {
        float v = c[g];
        if (relu) v = fmaxf(v, 0.0f);
        dst[(r0 + g) * RS + col] = v;
    }
}

__device__ __forceinline__ void gemm_step(const float* abuf, int rb, int k0,
                                          const float* wfrag, int kt, int lane,
                                          v8f& c0, v8f& c1) {
    v2f a  = lafrag(abuf, rb, k0, lane);
    v2f b0 = lbfrag(wfrag, kt, 0, lane);
    v2f b1 = lbfrag(wfrag, kt, 1, lane);
    c0 = wmma4(a, b0, c0);
    c1 = wmma4(a, b1, c1);
}

// rank-1 first encoder layer folded into A-fragment generation
__device__ __forceinline__ v2f encfrag(const float* xs, int rb, int k0, int lane,
                                       const float* W1, const float* b1, float scale) {
    int   c = k0 + ((lane >> 4) << 1);
    float x = xs[rb + (lane & 15)] * scale;
    v2f a;
    a.x = fmaxf(x * W1[c]     + b1[c],     0.0f);
    a.y = fmaxf(x * W1[c + 1] + b1[c + 1], 0.0f);
    return a;
}

// ---------------- norm = ||edges|| (deterministic two-stage) ----------------
__global__ void precnet_norm_partial(const float* __restrict__ edges, float* __restrict__ ws) {
    __shared__ float red[NTHREADS];
    int tid = threadIdx.x;
    float acc = 0.0f;
    for (int i = blockIdx.x * NTHREADS + tid; i < E_EDGES; i += 256 * NTHREADS) {
        float v = edges[i];
        acc += v * v;
    }
    red[tid] = acc;
    __syncthreads();
    for (int s = NTHREADS / 2; s > 0; s >>= 1) {
        if (tid < s) red[tid] += red[tid + s];
        __syncthreads();
    }
    if (tid == 0) ws[16 + blockIdx.x] = red[0];
}

__global__ void precnet_norm_final(float* __restrict__ ws) {
    __shared__ float red[NTHREADS];
    int tid = threadIdx.x;
    red[tid] = ws[16 + tid];
    __syncthreads();
    for (int s = NTHREADS / 2; s > 0; s >>= 1) {
        if (tid < s) red[tid] += red[tid + s];
        __syncthreads();
    }
    if (tid == 0) {
        float n = sqrtf(red[0]);
        ws[0] = n;
        ws[1] = 1.0f / n;
    }
}

// ---------------- fused 3-round MP-GNN over the ring ----------------
__global__ void precnet_fused(const float* __restrict__ nodesG, const float* __restrict__ edgesG,
                              const float* __restrict__ lhsE,
                              const float* __restrict__ neW1, const float* __restrict__ neB1,
                              const float* __restrict__ neW2, const float* __restrict__ neB2,
                              const float* __restrict__ eeW1, const float* __restrict__ eeB1,
                              const float* __restrict__ eeW2, const float* __restrict__ eeB2,
                              const float* __restrict__ mpWe, const float* __restrict__ mpBe,
                              const float* __restrict__ mpWn, const float* __restrict__ mpBn,
                              const float* __restrict__ edW1, const float* __restrict__ edB1,
                              const float* __restrict__ edW2, const float* __restrict__ edB2,
                              const float* __restrict__ ws, float* __restrict__ out) {
    extern __shared__ float lds[];
    const int tid  = threadIdx.x;
    const int lane = tid & 31;
    const int wave = tid >> 5;
    const int a0   = blockIdx.x * FAM - 2 + N_NODES;  // local row r -> global (a0+r)%N

    // ---- phase 0: weights -> packed B-fragments; vectors; raw input staging ----
    for (int i = tid; i < 8192; i += NTHREADS) {
        const float* g;
        int base, off = i;
        if (i < 1024)      { g = neW2; base = F_NEW2; }
        else if (i < 2048) { g = eeW2; base = F_EEW2; off = i - 1024; }
        else if (i < 5120) { g = mpWe; base = F_MPWE; off = i - 2048; }
        else if (i < 7168) { g = mpWn; base = F_MPWN; off = i - 5120; }
        else               { g = edW1; base = F_EDW1; off = i - 7168; }
        int e  = off & 1;
        int l  = (off >> 1) & 31;
        int f  = off >> 6;
        int kt = f >> 1, nt = f & 1;
        int row = (kt << 2) + e + ((l >> 4) << 1);
        int col = (nt << 4) + (l & 15);
        lds[base + off] = g[row * HID + col];
    }
    if (tid < 32) {
        lds[V_NEW1 + tid] = neW1[tid]; lds[V_NEB1 + tid] = neB1[tid];
        lds[V_NEB2 + tid] = neB2[tid]; lds[V_EEW1 + tid] = eeW1[tid];
        lds[V_EEB1 + tid] = eeB1[tid]; lds[V_EEB2 + tid] = eeB2[tid];
        lds[V_MPBE + tid] = mpBe[tid]; lds[V_MPBN + tid] = mpBn[tid];
        lds[V_EDB1 + tid] = edB1[tid]; lds[V_EDW2 + tid] = edW2[tid];
    }
    if (tid == 0) {
        lds[V_EDB2] = edB2[0];
        lds[V_NORM] = ws[0];
        lds[V_INV]  = ws[1];
    }
    for (int i = tid; i < 4 * MROWS; i += NTHREADS) {
        int t = i / MROWS, r = i - t * MROWS;
        int v = (a0 + r) % N_NODES;
        lds[X_O + i] = (t == 0) ? nodesG[v] : edgesG[(t - 1) * N_NODES + v];
    }
    __syncthreads();

    float* nb = lds + NBUF_O;
    const float inv = lds[V_INV];

    // ---- encoders: 5 node tiles + 15 edge tiles (20 wave jobs) ----
    for (int job = wave; job < 20; job += 8) {
        const float* xs; const float* W1; const float* b1;
        const float* wf; const float* b2; float* dst; float sc; int mb;
        if (job < 5) {
            xs = lds + X_O; W1 = lds + V_NEW1; b1 = lds + V_NEB1;
            wf = lds + F_NEW2; b2 = lds + V_NEB2; dst = nb; sc = 1.0f;
            mb = job * 16;
        } else {
            int t = (job - 5) / 5, m = (job - 5) % 5;
            xs = lds + X_O + (t + 1) * MROWS; W1 = lds + V_EEW1; b1 = lds + V_EEB1;
            wf = lds + F_EEW2; b2 = lds + V_EEB2; dst = lds + E_O + t * BUFSZ; sc = inv;
            mb = m * 16;
        }
        v8f c0 = cbias(b2, 0, lane), c1 = cbias(b2, 1, lane);
#pragma unroll
        for (int kt = 0; kt < 8; ++kt) {
            v2f a  = encfrag(xs, mb, kt * 4, lane, W1, b1, sc);
            v2f bb0 = lbfrag(wf, kt, 0, lane);
            v2f bb1 = lbfrag(wf, kt, 1, lane);
            c0 = wmma4(a, bb0, c0);
            c1 = wmma4(a, bb1, c1);
        }
        cstore(dst, mb, 0, lane, c0, false);
        cstore(dst, mb, 1, lane, c1, false);
    }
    __syncthreads();

    // ---- 3 message-passing rounds (last round: edge update only) ----
    for (int r = 0; r < 3; ++r) {
        // edge update: e = relu([e, n[snd], n[rcv]] @ mp_We + mp_be), in place
        for (int job = wave; job < 15; job += 8) {
            int t = job / 5, mb = (job % 5) * 16;
            float* et = lds + E_O + t * BUFSZ;
            int so = (t == 2) ? 1 : 0;   // sender row offset
            int ro = (t == 1) ? 1 : 0;   // receiver row offset
            v8f c0 = cbias(lds + V_MPBE, 0, lane), c1 = cbias(lds + V_MPBE, 1, lane);
#pragma unroll
            for (int kc = 0; kc < 8; ++kc)
                gemm_step(et, mb, kc * 4, lds + F_MPWE, kc, lane, c0, c1);
#pragma unroll
            for (int kc = 0; kc < 8; ++kc)
                gemm_step(nb, mb + so, kc * 4, lds + F_MPWE, 8 + kc, lane, c0, c1);
#pragma unroll
            for (int kc = 0; kc < 8; ++kc)
                gemm_step(nb, mb + ro, kc * 4, lds + F_MPWE, 16 + kc, lane, c0, c1);
            cstore(et, mb, 0, lane, c0, true);
            cstore(et, mb, 1, lane, c1, true);
        }
        __syncthreads();
        if (r == 2) break;

        // agg[v] = e0[v] + e1[v-1] + e2[v]   (ring gather, deterministic)
        for (int i = tid; i < MROWS * HID; i += NTHREADS) {
            int row = i >> 5, c = i & 31;
            float v = 0.0f;
            if (row > 0)
                v = lds[E_O + row * RS + c] + lds[E_O + BUFSZ + (row - 1) * RS + c] +
                    lds[E_O + 2 * BUFSZ + row * RS + c];
            lds[SC_O + row * RS + c] = v;
        }
        __syncthreads();

        // node update: n = relu([n, agg] @ mp_Wn + mp_bn), in place (tile-private rows)
        for (int job = wave; job < 5; job += 8) {
            int mb = job * 16;
            v8f c0 = cbias(lds + V_MPBN, 0, lane), c1 = cbias(lds + V_MPBN, 1, lane);
#pragma unroll
            for (int kc = 0; kc < 8; ++kc)
                gemm_step(nb, mb, kc * 4, lds + F_MPWN, kc, lane, c0, c1);
#pragma unroll
            for (int kc = 0; kc < 8; ++kc)
                gemm_step(lds + SC_O, mb, kc * 4, lds + F_MPWN, 8 + kc, lane, c0, c1);
            cstore(nb, mb, 0, lane, c0, true);
            cstore(nb, mb, 1, lane, c1, true);
        }
        __syncthreads();
    }

    // ---- bi-edge averaging over output families (local rows 2..65) ----
    for (int i = tid; i < FAM * HID; i += NTHREADS) {
        int f = 2 + (i >> 5), c = i & 31;
        float av = 0.5f * (lds[E_O + BUFSZ + f * RS + c] + lds[E_O + 2 * BUFSZ + f * RS + c]);
        lds[E_O + BUFSZ + f * RS + c]     = av;
        lds[E_O + 2 * BUFSZ + f * RS + c] = av;
    }
    __syncthreads();

    // ---- decoder layer 1 (types 1,2 only; self-edges get sqrt overwrite) ----
    for (int job = wave; job < 8; job += 8) {
        int t = 1 + (job >> 2), mb = 2 + (job & 3) * 16;
        float* et = lds + E_O + t * BUFSZ;
        v8f c0 = cbias(lds + V_EDB1, 0, lane), c1 = cbias(lds + V_EDB1, 1, lane);
#pragma unroll
        for (int kc = 0; kc < 8; ++kc)
            gemm_step(et, mb, kc * 4, lds + F_EDW1, kc, lane, c0, c1);
        cstore(et, mb, 0, lane, c0, true);
        cstore(et, mb, 1, lane, c1, true);
    }
    __syncthreads();

    // ---- decoder layer 2 (dot with ed_W2), masking, final writes ----
    const float norm = lds[V_NORM];
    for (int i = tid; i < 3 * FAM; i += NTHREADS) {
        int t  = i >> 6;
        int f2 = i & 63;
        int k  = blockIdx.x * FAM + f2;
        int fl = 2 + f2;
        float data; int rr, cc;
        if (t == 0) {
            data = sqrtf(lhsE[k]);  // diag overwrite: sqrt(lhs_edges[diag]); mask true
            rr = k; cc = k;
        } else {
            float acc = lds[V_EDB2];
            const float* h = lds + E_O + t * BUFSZ + fl * RS;
#pragma unroll
            for (int c = 0; c < HID; ++c) acc += h[c] * lds[V_EDW2 + c];
            float val = acc * norm;
            if (t == 1) {  // edge N+k: sender k, receiver (k+1)%N
                bool m = (k == N_NODES - 1);
                data = m ? val : 0.0f;
                rr = m ? k : 0;
                cc = m ? ((k + 1) % N_NODES) : 0;
            } else {       // edge 2N+k: sender (k+1)%N, receiver k
                bool m = (k != N_NODES - 1);
                data = m ? val : 0.0f;
                rr = m ? (k + 1) : 0;
                cc = m ? k : 0;
            }
        }
        int e = t * N_NODES + k;
        out[e]               = data;
        out[E_EDGES + e]     = (float)rr;   // indices < 2^24: exact in f32
        out[2 * E_EDGES + e] = (float)cc;
    }
}

extern "C" void kernel_launch(void* const* d_in, const int* in_sizes, int n_in,
                              void* d_out, int out_size, void* d_ws, size_t ws_size,
                              hipStream_t stream) {
    const float* nodes = (const float*)d_in[0];
    const float* edges = (const float*)d_in[1];
    const float* lhsE  = (const float*)d_in[5];
    const float* neW1  = (const float*)d_in[9];
    const float* neB1  = (const float*)d_in[10];
    const float* neW2  = (const float*)d_in[11];
    const float* neB2  = (const float*)d_in[12];
    const float* eeW1  = (const float*)d_in[13];
    const float* eeB1  = (const float*)d_in[14];
    const float* eeW2  = (const float*)d_in[15];
    const float* eeB2  = (const float*)d_in[16];
    const float* mpWe  = (const float*)d_in[17];
    const float* mpBe  = (const float*)d_in[18];
    const float* mpWn  = (const float*)d_in[19];
    const float* mpBn  = (const float*)d_in[20];
    const float* edW1  = (const float*)d_in[21];
    const float* edB1  = (const float*)d_in[22];
    const float* edW2  = (const float*)d_in[23];
    const float* edB2  = (const float*)d_in[24];
    float* ws  = (float*)d_ws;
    float* out = (float*)d_out;

    precnet_norm_partial<<<256, NTHREADS, 0, stream>>>(edges, ws);
    precnet_norm_final<<<1, NTHREADS, 0, stream>>>(ws);

    dim3 grid(N_NODES / FAM);
    size_t shmem = (size_t)LDS_FLOATS * sizeof(float);  // ~90 KB of 320 KB/WGP
    precnet_fused<<<grid, NTHREADS, shmem, stream>>>(
        nodes, edges, lhsE, neW1, neB1, neW2, neB2, eeW1, eeB1, eeW2, eeB2,
        mpWe, mpBe, mpWn, mpBn, edW1, edB1, edW2, edB2, ws, out);
}